// SocialMFModel_67585605370574
// MI455X (gfx1250) — compile-verified
//
#include <hip/hip_runtime.h>
#include <math.h>

typedef float v2f __attribute__((ext_vector_type(2)));
typedef float v8f __attribute__((ext_vector_type(8)));

#define EMB_D 64

// ---------------------------------------------------------------------------
// Wave-level sum reduction (wave32)
// ---------------------------------------------------------------------------
__device__ __forceinline__ float waveSum(float v) {
#pragma unroll
    for (int off = 16; off > 0; off >>= 1) v += __shfl_down(v, off, 32);
    return v;
}

// ---------------------------------------------------------------------------
// Edge dot-product scores via V_WMMA_F32_16X16X4_F32.
// One wave handles 16 edges. A = 16 user rows (16x4 f32 chunk),
// B = 16 item rows transposed (4x16 f32 chunk). Diagonal of the 16x16
// accumulator = the 16 dot products, accumulated over 16 K-chunks (D=64).
//
// ISA layouts (wave32):
//   A 16x4 : lanes 0-15 -> M=lane,  V0=K0,V1=K1 ; lanes 16-31 -> M=lane-16, V0=K2,V1=K3
//   B 4x16 : lanes 0-15 -> N=lane,  V0=K0,V1=K1 ; lanes 16-31 -> N=lane-16, V0=K2,V1=K3
// => each lane loads a contiguous float2 at row*64 + 4*kc + 2*(lane>>4).
//   C/D 16x16: VGPR j: lanes0-15 (M=j,N=lane); lanes16-31 (M=j+8,N=lane-16)
// => diag (j,j) at c[j] lane j ; diag (j+8,j+8) at c[j] lane j+24.
// ---------------------------------------------------------------------------
__global__ __launch_bounds__(128) void score_wmma_kernel(
    const float* __restrict__ user_emb,
    const float* __restrict__ item_emb,
    const int* __restrict__ u_idx,
    const int* __restrict__ i_idx,
    float* __restrict__ out,
    int ngroups)  // groups of 16 edges
{
    const int wave = blockIdx.x * (blockDim.x >> 5) + (threadIdx.x >> 5);
    if (wave >= ngroups) return;  // wave-uniform: EXEC stays all-ones inside

    const int lane = threadIdx.x & 31;
    const int r    = lane & 15;   // edge-in-group / matrix row
    const int hi   = lane >> 4;   // 0 -> K0,K1 ; 1 -> K2,K3
    const long base = (long)wave * 16;

    const int u  = u_idx[base + r];
    const int it = i_idx[base + r];
    const float* ua = user_emb + (size_t)u  * EMB_D + (hi << 1);
    const float* ia = item_emb + (size_t)it * EMB_D + (hi << 1);

    v8f c = {};
#pragma unroll
    for (int kc = 0; kc < 16; ++kc) {
        v2f a = *(const v2f*)(ua + (kc << 2));
        v2f b = *(const v2f*)(ia + (kc << 2));
        // 8 args: (neg_a, A, neg_b, B, c_mod, C, reuse_a, reuse_b)
        c = __builtin_amdgcn_wmma_f32_16x16x4_f32(
                false, a, false, b, (short)0, c, false, false);
    }

    // Extract the diagonal: lane e (e<16) ends up holding score for edge base+e.
    float score = 0.0f;
#pragma unroll
    for (int j = 0; j < 8; ++j) {
        float lo = __shfl(c[j], j, 32);       // element (j, j)
        float hh = __shfl(c[j], j + 24, 32);  // element (j+8, j+8)
        if (lane == j)     score = lo;
        if (lane == j + 8) score = hh;
    }
    if (lane < 16)
        __builtin_nontemporal_store(score, out + base + lane);  // streaming output
}

// Scalar tail (edges not covered by full 16-groups). One wave per edge.
__global__ void score_tail_kernel(
    const float* __restrict__ user_emb, const float* __restrict__ item_emb,
    const int* __restrict__ u_idx, const int* __restrict__ i_idx,
    float* __restrict__ out, int start, int n_edges)
{
    int t = blockIdx.x * blockDim.x + threadIdx.x;
    int e = start + (t >> 5);
    if (e >= n_edges) return;
    int lane = t & 31;
    const v2f a = *(const v2f*)(user_emb + (size_t)u_idx[e] * EMB_D + lane * 2);
    const v2f b = *(const v2f*)(item_emb + (size_t)i_idx[e] * EMB_D + lane * 2);
    float s = a[0] * b[0] + a[1] * b[1];
    s = waveSum(s);
    if (lane == 0) out[e] = s;
}

// ---------------------------------------------------------------------------
// Trust-edge scatter: ft[dst] += user_emb[src] ; deg[dst] += 1.
// One wave per edge, each lane owns 2 dims (float2 gather + 2 f32 atomics).
// ft (25.6 MB) is L2-resident -> atomics resolve in L2.
// ---------------------------------------------------------------------------
__global__ void trust_scatter_kernel(
    const float* __restrict__ user_emb,
    const int* __restrict__ src, const int* __restrict__ dst,
    float* __restrict__ ft, float* __restrict__ deg, int n_edges)
{
    long t = (long)blockIdx.x * blockDim.x + threadIdx.x;
    long e = t >> 5;
    if (e >= n_edges) return;
    int lane = (int)(t & 31);
    int s = src[e], d = dst[e];
    const v2f v = *(const v2f*)(user_emb + (size_t)s * EMB_D + lane * 2);
    float* base = ft + (size_t)d * EMB_D + lane * 2;
    atomicAdd(base + 0, v[0]);
    atomicAdd(base + 1, v[1]);
    if (lane == 0) atomicAdd(deg + d, 1.0f);
}

// ---------------------------------------------------------------------------
// Reduction over user table: accumulates
//   acc[0] += (ft/max(deg,1) - user_emb)^2   (link loss, squared)
//   acc[1] += user_emb^2                     (user reg, squared)
// ---------------------------------------------------------------------------
__global__ __launch_bounds__(256) void reduce_user_kernel(
    const float* __restrict__ user_emb,
    const float* __restrict__ ft, const float* __restrict__ deg,
    float* __restrict__ acc, int n)  // n = U * 64
{
    __shared__ float s0[8], s1[8];
    int idx = blockIdx.x * blockDim.x + threadIdx.x;
    float sl = 0.0f, su = 0.0f;
    if (idx < n) {
        float ue = user_emb[idx];
        int   row = idx >> 6;
        float dg  = fmaxf(deg[row], 1.0f);
        float dl  = ft[idx] / dg - ue;
        sl = dl * dl;
        su = ue * ue;
    }
    sl = waveSum(sl);
    su = waveSum(su);
    int w = threadIdx.x >> 5, lane = threadIdx.x & 31;
    if (lane == 0) { s0[w] = sl; s1[w] = su; }
    __syncthreads();
    if (threadIdx.x == 0) {
        float a = 0.0f, b = 0.0f;
#pragma unroll
        for (int i = 0; i < 8; ++i) { a += s0[i]; b += s1[i]; }
        atomicAdd(acc + 0, a);
        atomicAdd(acc + 1, b);
    }
}

// acc[2] += item_emb^2
__global__ __launch_bounds__(256) void reduce_item_kernel(
    const float* __restrict__ item_emb, float* __restrict__ acc, int n)
{
    __shared__ float s0[8];
    int idx = blockIdx.x * blockDim.x + threadIdx.x;
    float si = 0.0f;
    if (idx < n) { float v = item_emb[idx]; si = v * v; }
    si = waveSum(si);
    int w = threadIdx.x >> 5, lane = threadIdx.x & 31;
    if (lane == 0) s0[w] = si;
    __syncthreads();
    if (threadIdx.x == 0) {
        float a = 0.0f;
#pragma unroll
        for (int i = 0; i < 8; ++i) a += s0[i];
        atomicAdd(acc + 2, a);
    }
}

// Final scalars: reg_loss = 0.5*(||U||+||I||) ; link_loss = 0.45*||pred-U||
__global__ void finalize_kernel(const float* __restrict__ acc,
                                float* __restrict__ out_scalars)
{
    if (threadIdx.x == 0 && blockIdx.x == 0) {
        out_scalars[0] = 0.5f  * (sqrtf(acc[1]) + sqrtf(acc[2]));  // reg_loss
        out_scalars[1] = 0.45f * sqrtf(acc[0]);                     // link_loss
    }
}

// ---------------------------------------------------------------------------
extern "C" void kernel_launch(void* const* d_in, const int* in_sizes, int n_in,
                              void* d_out, int out_size, void* d_ws, size_t ws_size,
                              hipStream_t stream)
{
    const float* user_emb = (const float*)d_in[0];
    const float* item_emb = (const float*)d_in[1];
    const int* trust_src  = (const int*)d_in[2];
    const int* trust_dst  = (const int*)d_in[3];
    const int* pos_u      = (const int*)d_in[4];
    const int* pos_i      = (const int*)d_in[5];
    const int* neg_u      = (const int*)d_in[6];
    const int* neg_i      = (const int*)d_in[7];
    float* out = (float*)d_out;

    const int U       = in_sizes[0] / EMB_D;
    const int I       = in_sizes[1] / EMB_D;
    const int E_TRUST = in_sizes[2];
    const int E_RATE  = in_sizes[4];

    // Workspace layout (floats): ft[U*64] | deg[U] | acc[3]
    float* ft  = (float*)d_ws;
    float* deg = ft + (size_t)U * EMB_D;
    float* acc = deg + U;
    const size_t zero_bytes = ((size_t)U * EMB_D + U + 3) * sizeof(float);
    hipMemsetAsync(d_ws, 0, zero_bytes, stream);

    float* pos_out = out;                 // [E_RATE]
    float* neg_out = out + E_RATE;        // [E_RATE]
    float* scalars = out + 2L * E_RATE;   // [reg_loss, link_loss]

    // --- Scores via WMMA f32 16x16x4 (16 edges per wave) ---
    const int ngroups = E_RATE / 16;
    const int waves_per_block = 4;  // 128 threads
    const int nblk = (ngroups + waves_per_block - 1) / waves_per_block;
    if (ngroups > 0) {
        score_wmma_kernel<<<nblk, 128, 0, stream>>>(user_emb, item_emb,
                                                    pos_u, pos_i, pos_out, ngroups);
        score_wmma_kernel<<<nblk, 128, 0, stream>>>(user_emb, item_emb,
                                                    neg_u, neg_i, neg_out, ngroups);
    }
    const int tail_start = ngroups * 16;
    const int tail = E_RATE - tail_start;
    if (tail > 0) {
        int tthreads = tail * 32;
        int tblk = (tthreads + 255) / 256;
        score_tail_kernel<<<tblk, 256, 0, stream>>>(user_emb, item_emb,
                                                    pos_u, pos_i, pos_out, tail_start, E_RATE);
        score_tail_kernel<<<tblk, 256, 0, stream>>>(user_emb, item_emb,
                                                    neg_u, neg_i, neg_out, tail_start, E_RATE);
    }

    // --- Trust scatter (segment-sum + degree) ---
    {
        long threads = (long)E_TRUST * 32;
        int blk = (int)((threads + 255) / 256);
        trust_scatter_kernel<<<blk, 256, 0, stream>>>(user_emb, trust_src, trust_dst,
                                                      ft, deg, E_TRUST);
    }

    // --- Norm reductions ---
    {
        int n = U * EMB_D;
        int blk = (n + 255) / 256;
        reduce_user_kernel<<<blk, 256, 0, stream>>>(user_emb, ft, deg, acc, n);
    }
    {
        int n = I * EMB_D;
        int blk = (n + 255) / 256;
        reduce_item_kernel<<<blk, 256, 0, stream>>>(item_emb, acc, n);
    }
    finalize_kernel<<<1, 32, 0, stream>>>(acc, scalars);
}